// Attn_6219112645241
// MI455X (gfx1250) — compile-verified
//
#include <hip/hip_runtime.h>
#include <math.h>

#define S_DIM 8192
#define H_DIM 2048

typedef __attribute__((ext_vector_type(2))) float v2f;
typedef __attribute__((ext_vector_type(4))) float v4f;
typedef __attribute__((ext_vector_type(8))) float v8f;

// ---------------------------------------------------------------------------
// Phase 1: u = h^T * W  via V_WMMA_F32_16X16X4_F32.
// A (16x4): row 0 = h[k0..k0+3], rows 1..15 zero.
//   A layout: lanes 0-15 hold M=0..15; VGPR0 = {K=0 | K=2}, VGPR1 = {K=1 | K=3}
//   -> only lane 0 (M=0,K=0/1) and lane 16 (M=0,K=2/3) carry h.
// A-fragment built branch-free with precomputed 0/1 lane masks so the
// compiler cannot sink loads into divergent EXEC blocks.
// B (4x16): rows of W striped across lanes; VGPR0 = {K=0 row | K=2 row},
//   VGPR1 = {K=1 row | K=3 row} -> 64B-contiguous bursts per half-wave.
// C row 0 (VGPR0, lanes 0-15) accumulates u[j0..j0+15].
// K split 4 ways for memory parallelism; partials summed in phase 2.
// ---------------------------------------------------------------------------
#define KSPLIT 4
__global__ __launch_bounds__(256) void attn_phase1_uW_wmma(
    const float* __restrict__ h, const float* __restrict__ W,
    float* __restrict__ u_partial /* [KSPLIT][H_DIM] */) {
  const int lane   = threadIdx.x & 31;
  const int waveId = (blockIdx.x * blockDim.x + threadIdx.x) >> 5;  // 0..511
  const int tile   = waveId & 127;   // 128 tiles of 16 columns
  const int kchunk = waveId >> 7;    // 0..3
  const int j0     = tile * 16;
  const int n      = lane & 15;
  const int hi     = lane >> 4;      // 0: lanes 0-15, 1: lanes 16-31
  const int kbeg   = kchunk * (H_DIM / KSPLIT);
  const int kend   = kbeg + (H_DIM / KSPLIT);

  // branch-free lane masks for the A fragment
  const float m_lo = (lane == 0)  ? 1.0f : 0.0f;
  const float m_hi = (lane == 16) ? 1.0f : 0.0f;

  const v4f* __restrict__ h4 = (const v4f*)h;

  v8f c = {};
  for (int k0 = kbeg; k0 < kend; k0 += 4) {
    // W tile: per-lane row (k0 + hi*2) and (k0 + hi*2 + 1), column j0+n
    const float* wp = W + (size_t)(k0 + hi * 2) * H_DIM + j0 + n;
    v2f b;
    b.x = wp[0];
    b.y = wp[H_DIM];

    // uniform 128-bit load of h[k0..k0+3]
    v4f hv = h4[k0 >> 2];
    v2f a;
    a.x = fmaf(m_lo, hv.x, m_hi * hv.z);
    a.y = fmaf(m_lo, hv.y, m_hi * hv.w);

    c = __builtin_amdgcn_wmma_f32_16x16x4_f32(false, a, false, b,
                                              (short)0, c, false, false);
  }
  if (lane < 16) u_partial[kchunk * H_DIM + j0 + lane] = c[0];
}

// ---------------------------------------------------------------------------
// Phase 2: energies[s] = dot(enc[s,:], u).  One wave32 per row; u summed from
// the KSPLIT partials into LDS once per block; enc streamed once as 128-bit
// vectors with non-temporal hint (64 MB single-pass, zero reuse).
// ---------------------------------------------------------------------------
__global__ __launch_bounds__(256) void attn_phase2_energies(
    const float* __restrict__ enc, const float* __restrict__ u_partial,
    float* __restrict__ energies) {
  __shared__ __align__(16) float u_lds[H_DIM];
  for (int i = threadIdx.x; i < H_DIM; i += 256) {
    u_lds[i] = u_partial[i] + u_partial[H_DIM + i] +
               u_partial[2 * H_DIM + i] + u_partial[3 * H_DIM + i];
  }
  __syncthreads();

  const int lane = threadIdx.x & 31;
  const int wave = threadIdx.x >> 5;
  const int s    = blockIdx.x * 8 + wave;          // 1024 blocks * 8 waves = 8192

  const v4f* __restrict__ enc4 = (const v4f*)enc + (size_t)s * (H_DIM / 4);
  const v4f* u4 = (const v4f*)u_lds;

  float acc = 0.0f;
#pragma unroll
  for (int c = 0; c < H_DIM / 4 / 32; ++c) {       // 16 chunks
    v4f v  = __builtin_nontemporal_load(&enc4[c * 32 + lane]);
    v4f uu = u4[c * 32 + lane];
    acc = fmaf(v.x, uu.x, acc);
    acc = fmaf(v.y, uu.y, acc);
    acc = fmaf(v.z, uu.z, acc);
    acc = fmaf(v.w, uu.w, acc);
  }
#pragma unroll
  for (int off = 16; off > 0; off >>= 1) acc += __shfl_xor(acc, off, 32);
  if (lane == 0) energies[s] = acc;
}

// ---------------------------------------------------------------------------
// Phase 3: softmax over S=8192 (32 KB — single block).  The reference's bias
// adds the constant b.h to every energy; softmax is shift-invariant -> b
// dropped.
// ---------------------------------------------------------------------------
__global__ __launch_bounds__(1024) void attn_phase3_softmax(
    const float* __restrict__ energies, float* __restrict__ out) {
  __shared__ float red[1024];
  float vals[8];
  float m = -__builtin_inff();
#pragma unroll
  for (int i = 0; i < 8; ++i) {
    vals[i] = energies[threadIdx.x + i * 1024];
    m = fmaxf(m, vals[i]);
  }
  red[threadIdx.x] = m;
  __syncthreads();
  for (int s2 = 512; s2 > 0; s2 >>= 1) {
    if (threadIdx.x < s2)
      red[threadIdx.x] = fmaxf(red[threadIdx.x], red[threadIdx.x + s2]);
    __syncthreads();
  }
  m = red[0];
  __syncthreads();

  float e[8];
  float sum = 0.0f;
#pragma unroll
  for (int i = 0; i < 8; ++i) {
    e[i] = expf(vals[i] - m);
    sum += e[i];
  }
  red[threadIdx.x] = sum;
  __syncthreads();
  for (int s2 = 512; s2 > 0; s2 >>= 1) {
    if (threadIdx.x < s2) red[threadIdx.x] += red[threadIdx.x + s2];
    __syncthreads();
  }
  const float inv = 1.0f / red[0];
#pragma unroll
  for (int i = 0; i < 8; ++i) out[threadIdx.x + i * 1024] = e[i] * inv;
}

// ---------------------------------------------------------------------------
extern "C" void kernel_launch(void* const* d_in, const int* in_sizes, int n_in,
                              void* d_out, int out_size, void* d_ws, size_t ws_size,
                              hipStream_t stream) {
  const float* hidden = (const float*)d_in[0];       // (1, H)
  const float* enc    = (const float*)d_in[1];       // (S, 1, H)
  const float* W      = (const float*)d_in[2];       // (H, H)
  // d_in[3] = b: softmax invariant to the uniform +b.h shift -> unused.
  float* out = (float*)d_out;                        // (1,1,S) = 8192 f32

  float* ws        = (float*)d_ws;
  float* u_partial = ws;                   // KSPLIT * H_DIM = 8192 floats
  float* energies  = ws + KSPLIT * H_DIM;  // 8192 floats

  attn_phase1_uW_wmma<<<64, 256, 0, stream>>>(hidden, W, u_partial);
  attn_phase2_energies<<<S_DIM / 8, 256, 0, stream>>>(enc, u_partial, energies);
  attn_phase3_softmax<<<1, 1024, 0, stream>>>(energies, out);
}